// skipgram_12472585028178
// MI455X (gfx1250) — compile-verified
//
#include <hip/hip_runtime.h>

// Skip-gram negative-sampling loss on MI455X (gfx1250, wave32).
// Memory-bound gather problem (~184 MB @ 23.3 TB/s ~= 8 us floor); dot
// products are routed through V_WMMA_F32_16X16X4_F32 (exact fp32 FMA).

typedef __attribute__((ext_vector_type(2))) float v2f;
typedef __attribute__((ext_vector_type(8))) float v8f;

#define DIM   128
#define NEG   20
#define BATCH 16384
#define TILES_PER_BLOCK 8   // 8 waves * 16 rows = 128 rows per block

__global__ void sg_zero_out(float* out) { out[0] = 0.0f; }

__device__ __forceinline__ float log_sigmoid(float x) {
    // stable: logσ(x) = min(x,0) - log1p(exp(-|x|))
    return fminf(x, 0.0f) - log1pf(__expf(-fabsf(x)));
}

__global__ __launch_bounds__(256)
void sg_loss_wmma(const int* __restrict__ u_pos,
                  const int* __restrict__ v_pos,
                  const int* __restrict__ v_neg,
                  const float* __restrict__ u_emb,
                  const float* __restrict__ v_emb,
                  float* __restrict__ out)
{
    const int wave = threadIdx.x >> 5;
    const int lane = threadIdx.x & 31;
    const int tile = blockIdx.x * TILES_PER_BLOCK + wave;   // 16 batch rows
    const int half = lane >> 4;                             // K sub-slot
    const int r    = lane & 15;                             // row within tile
    const int row  = tile * 16 + r;

    // Gathered row base offsets (fit in 32 bits: 100000*128 floats < 2^31).
    const unsigned uoff = (unsigned)u_pos[row] * DIM;
    const unsigned voff = (unsigned)v_pos[row] * DIM;

    unsigned noff[NEG];
    const int* np = v_neg + (size_t)row * NEG;
#pragma unroll
    for (int n = 0; n < NEG; ++n)
        noff[n] = (unsigned)np[n] * DIM;

    v8f cpos = {};   // D[m][n] = dot(u_m, v_n)        -> need diagonal
    v8f cneg = {};   // D[m][n] = dot(u_m, sum_neg_n)  -> need diagonal

    // K loop: 32 steps of K=4. Per step each lane supplies a float2 of A
    // (U[r][col..col+1]) and a float2 of B (V[r][col..col+1]); B for the
    // negative accumulator is the on-the-fly sum of the 20 negative rows.
#pragma unroll 2
    for (int k = 0; k < DIM; k += 4) {
        const int col = k + 2 * half;
        v2f a  = *(const v2f*)(u_emb + uoff + col);
        v2f bp = *(const v2f*)(v_emb + voff + col);
        v2f bn = {0.0f, 0.0f};
#pragma unroll
        for (int n = 0; n < NEG; ++n)
            bn += *(const v2f*)(v_emb + noff[n] + col);

        // 8 args: (neg_a, A, neg_b, B, c_mod, C, reuse_a, reuse_b)
        cpos = __builtin_amdgcn_wmma_f32_16x16x4_f32(
            false, a, false, bp, (short)0, cpos, false, false);
        cneg = __builtin_amdgcn_wmma_f32_16x16x4_f32(
            false, a, false, bn, (short)0, cneg, false, false);
    }

    // Diagonal extraction from the 16x16 f32 C/D layout:
    //   row m<8  -> lane m,     VGPR m
    //   row m>=8 -> lane m+16,  VGPR m-8
    int sel = -1;
    if (lane < 8)        sel = lane;        // rows 0..7
    else if (lane >= 24) sel = lane - 24;   // rows 8..15
    float spos = 0.0f, sneg = 0.0f;
#pragma unroll
    for (int i = 0; i < 8; ++i) {
        if (sel == i) { spos = cpos[i]; sneg = cneg[i]; }
    }

    float contrib = 0.0f;
    if (sel >= 0)
        contrib = -(log_sigmoid(spos) + log_sigmoid(-sneg)) * (1.0f / BATCH);

    // wave32 reduction
#pragma unroll
    for (int off = 16; off > 0; off >>= 1)
        contrib += __shfl_xor(contrib, off, 32);

    if (lane == 0)
        atomicAdd(out, contrib);
}

extern "C" void kernel_launch(void* const* d_in, const int* in_sizes, int n_in,
                              void* d_out, int out_size, void* d_ws, size_t ws_size,
                              hipStream_t stream) {
    (void)in_sizes; (void)n_in; (void)out_size; (void)d_ws; (void)ws_size;
    const int*   u_pos = (const int*)d_in[0];
    const int*   v_pos = (const int*)d_in[1];
    const int*   v_neg = (const int*)d_in[2];
    // d_in[3] = batch_size scalar (compile-time constant BATCH here)
    const float* u_emb = (const float*)d_in[4];
    const float* v_emb = (const float*)d_in[5];
    float* out = (float*)d_out;

    sg_zero_out<<<1, 1, 0, stream>>>(out);

    const int tiles  = BATCH / 16;                 // 1024
    const int blocks = tiles / TILES_PER_BLOCK;    // 128
    sg_loss_wmma<<<blocks, 256, 0, stream>>>(u_pos, v_pos, v_neg,
                                             u_emb, v_emb, out);
}